// Net1_36335423324472
// MI455X (gfx1250) — compile-verified
//
#include <hip/hip_runtime.h>
#include <hip/hip_bf16.h>

typedef float v2f __attribute__((ext_vector_type(2)));
typedef float v8f __attribute__((ext_vector_type(8)));

#define NGRAPHS 512
#define FDIM 32
#define CATDIM 160

// ---------------------------------------------------------------------------
// Edge scatter: agg[dst] += ew * h[src]   (one thread per (edge, 4-feature))
// h has row stride hstride (32 for layer0 input x, 160 for hcat slices).
// ---------------------------------------------------------------------------
__global__ void edge_scatter_kernel(const float* __restrict__ h, int hstride,
                                    const int* __restrict__ src,
                                    const int* __restrict__ dst,
                                    const float* __restrict__ ew,
                                    float* __restrict__ agg, int total) {
  int tid = blockIdx.x * blockDim.x + threadIdx.x;
  if (tid >= total) return;                  // total = E * 8
  int e  = tid >> 3;
  int f4 = (tid & 7) * 4;
  int s = src[e];
  int d = dst[e];
  float w = ew[e];
  const float4 xv = *(const float4*)(h + (long long)s * hstride + f4);
  float* ap = agg + (long long)d * FDIM + f4;
  atomicAdd(ap + 0, w * xv.x);
  atomicAdd(ap + 1, w * xv.y);
  atomicAdd(ap + 2, w * xv.z);
  atomicAdd(ap + 3, w * xv.w);
}

// ---------------------------------------------------------------------------
// Node update via fp32 WMMA 16x16x4:
//   out = relu(agg @ w_rel + b_rel + hprev @ w_root)
// Written into concat buffer hcat at column offset outoff (row stride 160).
// One wave handles 16 nodes x 32 output cols (two 16-col tiles).
// ---------------------------------------------------------------------------
__global__ void node_gemm_relu_kernel(const float* __restrict__ agg,
                                      const float* __restrict__ hprev, int hstride,
                                      const float* __restrict__ w_rel,
                                      const float* __restrict__ w_root,
                                      const float* __restrict__ bias,
                                      float* __restrict__ hcat, int outoff,
                                      int N) {
  int wave = threadIdx.x >> 5;
  int lane = threadIdx.x & 31;
  int row0 = (blockIdx.x * (blockDim.x >> 5) + wave) * 16;
  if (row0 >= N) return;                     // wave-uniform: EXEC stays all-ones

  int m   = lane & 15;                       // A row-in-tile / B,D column
  int kh  = (lane >> 4) * 2;                 // K sub-offset: lanes 16-31 hold K+2,K+3
  long long arow = (long long)(row0 + m);

  for (int ct = 0; ct < 2; ++ct) {
    v8f acc = {};
    const float* wr = w_rel  + ct * 16;
    const float* wo = w_root + ct * 16;
#pragma unroll
    for (int k = 0; k < FDIM; k += 4) {      // agg @ w_rel
      v2f a, b;
      a.x = agg[arow * FDIM + k + kh];
      a.y = agg[arow * FDIM + k + kh + 1];
      b.x = wr[(k + kh) * FDIM + m];
      b.y = wr[(k + kh + 1) * FDIM + m];
      acc = __builtin_amdgcn_wmma_f32_16x16x4_f32(false, a, false, b,
                                                  (short)0, acc, false, false);
    }
#pragma unroll
    for (int k = 0; k < FDIM; k += 4) {      // hprev @ w_root
      v2f a, b;
      a.x = hprev[arow * hstride + k + kh];
      a.y = hprev[arow * hstride + k + kh + 1];
      b.x = wo[(k + kh) * FDIM + m];
      b.y = wo[(k + kh + 1) * FDIM + m];
      acc = __builtin_amdgcn_wmma_f32_16x16x4_f32(false, a, false, b,
                                                  (short)0, acc, false, false);
    }
    float bv = bias[ct * 16 + m];
    int colBase = outoff + ct * 16;
#pragma unroll
    for (int r = 0; r < 8; ++r) {
      int orow = row0 + r + (lane >> 4) * 8; // D: vgpr r, lanes>=16 -> M=r+8
      float v = acc[r] + bv;
      hcat[(long long)orow * CATDIM + colBase + m] = v > 0.f ? v : 0.f;
    }
  }
}

// ---------------------------------------------------------------------------
// Graph pooling: pooled[batch[n]] += hcat[n]  (one thread per (node, 4-feat))
// ---------------------------------------------------------------------------
__global__ void pool_kernel(const float* __restrict__ hcat,
                            const int* __restrict__ batch,
                            float* __restrict__ pooled, int total) {
  int tid = blockIdx.x * blockDim.x + threadIdx.x;
  if (tid >= total) return;                  // total = N * 40
  int n  = tid / 40;
  int c4 = (tid % 40) * 4;
  int g = batch[n];
  const float4 v = *(const float4*)(hcat + (long long)n * CATDIM + c4);
  float* pp = pooled + (long long)g * CATDIM + c4;
  atomicAdd(pp + 0, v.x);
  atomicAdd(pp + 1, v.y);
  atomicAdd(pp + 2, v.z);
  atomicAdd(pp + 3, v.w);
}

// ---------------------------------------------------------------------------
// FC1 via fp32 WMMA: z = relu(pooled @ w_fc1 + b_fc1), K = 160
// ---------------------------------------------------------------------------
__global__ void fc1_wmma_kernel(const float* __restrict__ pooled,
                                const float* __restrict__ w_fc1,
                                const float* __restrict__ b_fc1,
                                float* __restrict__ z) {
  int wave = threadIdx.x >> 5;
  int lane = threadIdx.x & 31;
  int row0 = (blockIdx.x * (blockDim.x >> 5) + wave) * 16;
  if (row0 >= NGRAPHS) return;

  int m  = lane & 15;
  int kh = (lane >> 4) * 2;
  long long arow = (long long)(row0 + m);

  for (int ct = 0; ct < 2; ++ct) {
    v8f acc = {};
    const float* wp = w_fc1 + ct * 16;
    for (int k = 0; k < CATDIM; k += 4) {
      v2f a, b;
      a.x = pooled[arow * CATDIM + k + kh];
      a.y = pooled[arow * CATDIM + k + kh + 1];
      b.x = wp[(k + kh) * FDIM + m];
      b.y = wp[(k + kh + 1) * FDIM + m];
      acc = __builtin_amdgcn_wmma_f32_16x16x4_f32(false, a, false, b,
                                                  (short)0, acc, false, false);
    }
    float bv = b_fc1[ct * 16 + m];
#pragma unroll
    for (int r = 0; r < 8; ++r) {
      int orow = row0 + r + (lane >> 4) * 8;
      float v = acc[r] + bv;
      z[orow * FDIM + ct * 16 + m] = v > 0.f ? v : 0.f;
    }
  }
}

// ---------------------------------------------------------------------------
// FC2 + log_softmax: one thread per graph.
// ---------------------------------------------------------------------------
__global__ void fc2_logsoftmax_kernel(const float* __restrict__ z,
                                      const float* __restrict__ w_fc2,
                                      const float* __restrict__ b_fc2,
                                      float* __restrict__ out) {
  int g = blockIdx.x * blockDim.x + threadIdx.x;
  if (g >= NGRAPHS) return;
  float l0 = b_fc2[0], l1 = b_fc2[1];
  const float* zr = z + g * FDIM;
#pragma unroll
  for (int i = 0; i < FDIM; ++i) {
    float zi = zr[i];
    l0 += zi * w_fc2[i * 2 + 0];
    l1 += zi * w_fc2[i * 2 + 1];
  }
  float mx = l0 > l1 ? l0 : l1;
  float lse = mx + __logf(__expf(l0 - mx) + __expf(l1 - mx));
  out[g * 2 + 0] = l0 - lse;
  out[g * 2 + 1] = l1 - lse;
}

extern "C" void kernel_launch(void* const* d_in, const int* in_sizes, int n_in,
                              void* d_out, int out_size, void* d_ws, size_t ws_size,
                              hipStream_t stream) {
  const float* x       = (const float*)d_in[0];
  const int*   eidx    = (const int*)d_in[1];     // [2, E]
  const int*   batch   = (const int*)d_in[2];     // [N]
  const float* ew      = (const float*)d_in[3];   // [E]
  const float* w_rel1  = (const float*)d_in[4];
  const float* b_rel1  = (const float*)d_in[5];
  const float* w_root1 = (const float*)d_in[6];
  const float* w_rel_r = (const float*)d_in[7];   // [4,32,32]
  const float* b_rel_r = (const float*)d_in[8];   // [4,32]
  const float* w_root_r= (const float*)d_in[9];   // [4,32,32]
  const float* w_fc1   = (const float*)d_in[10];  // [160,32]
  const float* b_fc1   = (const float*)d_in[11];  // [32]
  const float* w_fc2   = (const float*)d_in[12];  // [32,2]
  const float* b_fc2   = (const float*)d_in[13];  // [2]

  const int N = in_sizes[2];
  const int E = in_sizes[3];
  const int* src = eidx;
  const int* dst = eidx + E;

  // Workspace layout (floats): hcat[N*160] | agg[N*32] | pooled[512*160] | z[512*32]
  float* hcat   = (float*)d_ws;
  float* agg    = hcat + (long long)N * CATDIM;
  float* pooled = agg + (long long)N * FDIM;
  float* z      = pooled + NGRAPHS * CATDIM;

  const int edgeTotal = E * 8;
  const int edgeBlocks = (edgeTotal + 255) / 256;
  const int nodeWaves = (N + 15) / 16;
  const int nodeBlocks = (nodeWaves + 3) / 4;    // 4 waves / 128-thread block

  for (int l = 0; l < 5; ++l) {
    const float* hprev;
    int hstride;
    const float *wr, *br, *wo;
    if (l == 0) {
      hprev = x; hstride = FDIM;
      wr = w_rel1; br = b_rel1; wo = w_root1;
    } else {
      hprev = hcat + (l - 1) * FDIM; hstride = CATDIM;
      wr = w_rel_r + (l - 1) * FDIM * FDIM;
      br = b_rel_r + (l - 1) * FDIM;
      wo = w_root_r + (l - 1) * FDIM * FDIM;
    }
    hipMemsetAsync(agg, 0, (size_t)N * FDIM * sizeof(float), stream);
    edge_scatter_kernel<<<edgeBlocks, 256, 0, stream>>>(hprev, hstride, src, dst,
                                                        ew, agg, edgeTotal);
    node_gemm_relu_kernel<<<nodeBlocks, 128, 0, stream>>>(agg, hprev, hstride,
                                                          wr, wo, br,
                                                          hcat, l * FDIM, N);
  }

  hipMemsetAsync(pooled, 0, (size_t)NGRAPHS * CATDIM * sizeof(float), stream);
  const int poolTotal = N * 40;
  pool_kernel<<<(poolTotal + 255) / 256, 256, 0, stream>>>(hcat, batch, pooled,
                                                           poolTotal);

  fc1_wmma_kernel<<<(NGRAPHS / 16 + 3) / 4, 128, 0, stream>>>(pooled, w_fc1,
                                                              b_fc1, z);
  fc2_logsoftmax_kernel<<<(NGRAPHS + 255) / 256, 256, 0, stream>>>(z, w_fc2,
                                                                   b_fc2,
                                                                   (float*)d_out);
}